// MultimodalDenoisingEncoder_38920993636928
// MI455X (gfx1250) — compile-verified
//
#include <hip/hip_runtime.h>
#include <hip/hip_bf16.h>
#include <math.h>

// ---------------------------------------------------------------------------
// MultimodalDenoisingEncoder for MI455X (gfx1250, wave32)
// B=64, S=128, N=576, H=768, N_HEAD=12, HD=64, k_strong=403, m_weak=173
//
// Score stage refactored algebraically (43.5 GFLOP -> 0.68 GFLOP, memory
// bound at ~5us of the 23.3 TB/s HBM). The cosine-sim GEMM (6.85 GFLOP f32)
// runs on V_WMMA_F32_16X16X4_F32 with async global->LDS double buffering.
// ---------------------------------------------------------------------------

typedef __attribute__((ext_vector_type(2))) float v2f;
typedef __attribute__((ext_vector_type(8))) float v8f;
typedef int v4i __attribute__((vector_size(4 * sizeof(int))));  // matches builtin proto

#define BDIM      64
#define NTOK      576
#define HDIM      768
#define NHEAD     12
#define KSTRONG   403
#define MWEAK     173
#define MTILES    11      // ceil(173/16)
#define KTILES    26      // ceil(403/16)
#define LDSTRIDE  772     // 768 + 4 pad -> conflict-free across 64 banks

#define GLOBAL_AS __attribute__((address_space(1)))
#define LDS_AS    __attribute__((address_space(3)))

#if defined(__has_builtin)
#if __has_builtin(__builtin_amdgcn_global_load_async_to_lds_b128)
#define HAVE_ASYNC_LDS 1
#else
#define HAVE_ASYNC_LDS 0
#endif
#else
#define HAVE_ASYNC_LDS 0
#endif

__device__ __forceinline__ void wait_async_le96() {
#if HAVE_ASYNC_LDS
#if __has_builtin(__builtin_amdgcn_s_wait_asynccnt)
  __builtin_amdgcn_s_wait_asynccnt(96);
#else
  asm volatile("s_wait_asynccnt 0x60" ::: "memory");
#endif
#endif
}
__device__ __forceinline__ void wait_async_0() {
#if HAVE_ASYNC_LDS
#if __has_builtin(__builtin_amdgcn_s_wait_asynccnt)
  __builtin_amdgcn_s_wait_asynccnt(0);
#else
  asm volatile("s_wait_asynccnt 0x0" ::: "memory");
#endif
#endif
}

// ---------------------------------------------------------------------------
// K1: u[a,b,h] = sum_d w_kx[a,h,d] * qx[a,b,d],  qx[a,b,d]=sum_h q[b,h]*w_qx[a,h,d]
// grid (12,64), block 64
// ---------------------------------------------------------------------------
__global__ void k_compute_u(const float* __restrict__ text,
                            const float* __restrict__ wqx,
                            const float* __restrict__ wkx,
                            float* __restrict__ u) {
  const int a = blockIdx.x;
  const int b = blockIdx.y;
  const int d = threadIdx.x;                 // 0..63
  __shared__ float qx[64];

  const float* q  = text + (size_t)b * 128 * HDIM;  // text[:,0,:]
  const float* wq = wqx + (size_t)a * HDIM * 64;
  float acc = 0.f;
  for (int h = 0; h < HDIM; ++h)
    acc = fmaf(q[h], wq[h * 64 + d], acc);
  qx[d] = acc;
  __syncthreads();

  const float* wk = wkx + (size_t)a * HDIM * 64;
  float* urow = u + ((size_t)a * BDIM + b) * HDIM;
  for (int h = d; h < HDIM; h += 64) {
    const float* row = wk + (size_t)h * 64;
    float s = 0.f;
#pragma unroll 16
    for (int dd = 0; dd < 64; ++dd) s = fmaf(row[dd], qx[dd], s);
    urow[h] = s;
  }
}

// ---------------------------------------------------------------------------
// K2: pre-softmax scores S[a,b,n] = image[b,n,:].u[a,b,:]/8  and per-row
//     inverse L2 norms of image. One pass over image (memory bound).
// grid 64, block 256
// ---------------------------------------------------------------------------
__global__ void k_score_rnorm(const float* __restrict__ image,
                              const float* __restrict__ u,
                              float* __restrict__ P,
                              float* __restrict__ rnorm) {
  const int b = blockIdx.x;
  __shared__ __align__(16) float us[NHEAD * HDIM];
  for (int i = threadIdx.x; i < NHEAD * HDIM; i += blockDim.x) {
    int a = i / HDIM, h = i % HDIM;
    us[i] = u[((size_t)a * BDIM + b) * HDIM + h];
  }
  __syncthreads();

  for (int n = threadIdx.x; n < NTOK; n += blockDim.x) {
    const float4* row = (const float4*)(image + ((size_t)b * NTOK + n) * HDIM);
    float acc[NHEAD];
#pragma unroll
    for (int a = 0; a < NHEAD; ++a) acc[a] = 0.f;
    float ss = 0.f;
    for (int h4 = 0; h4 < HDIM / 4; ++h4) {
      const float4 x = row[h4];
      ss = fmaf(x.x, x.x, fmaf(x.y, x.y, fmaf(x.z, x.z, fmaf(x.w, x.w, ss))));
#pragma unroll
      for (int a = 0; a < NHEAD; ++a) {
        const float4 w = *(const float4*)&us[a * HDIM + h4 * 4];
        acc[a] = fmaf(x.x, w.x, fmaf(x.y, w.y, fmaf(x.z, w.z, fmaf(x.w, w.w, acc[a]))));
      }
    }
#pragma unroll
    for (int a = 0; a < NHEAD; ++a)
      P[((size_t)a * BDIM + b) * NTOK + n] = acc[a] * 0.125f;  // /sqrt(64)
    rnorm[b * NTOK + n] = 1.0f / fmaxf(sqrtf(ss), 1e-12f);
  }
}

// ---------------------------------------------------------------------------
// K3: softmax over n per flat row f = a*64+b. grid 768, block 256
// ---------------------------------------------------------------------------
__global__ void k_softmax(float* __restrict__ P) {
  float* row = P + (size_t)blockIdx.x * NTOK;
  __shared__ float red[256];
  const int t = threadIdx.x;

  float m = -INFINITY;
  for (int n = t; n < NTOK; n += 256) m = fmaxf(m, row[n]);
  red[t] = m; __syncthreads();
  for (int s = 128; s > 0; s >>= 1) {
    if (t < s) red[t] = fmaxf(red[t], red[t + s]);
    __syncthreads();
  }
  m = red[0]; __syncthreads();

  float sum = 0.f;
  for (int n = t; n < NTOK; n += 256) {
    float e = expf(row[n] - m);
    row[n] = e; sum += e;
  }
  red[t] = sum; __syncthreads();
  for (int s = 128; s > 0; s >>= 1) {
    if (t < s) red[t] += red[t + s];
    __syncthreads();
  }
  const float inv = 1.0f / red[0];
  for (int n = t; n < NTOK; n += 256) row[n] *= inv;
}

// ---------------------------------------------------------------------------
// K4: replicate the reference's reshape-mean exactly:
//     scores[b',n] = (1/12) * sum_j probs_flat[b'*12+j][n], flat f = a*64+b.
// grid 64, block 576
// ---------------------------------------------------------------------------
__global__ void k_mean(const float* __restrict__ P, float* __restrict__ scores) {
  const int bp = blockIdx.x;
  for (int n = threadIdx.x; n < NTOK; n += blockDim.x) {
    float s = 0.f;
#pragma unroll
    for (int j = 0; j < NHEAD; ++j)
      s += P[((size_t)bp * NHEAD + j) * NTOK + n];
    scores[bp * NTOK + n] = s * (1.0f / 12.0f);
  }
}

// ---------------------------------------------------------------------------
// K5: stable rank-based top-k (matches lax.top_k ordering incl. tie-break).
// grid 64, block 576
// ---------------------------------------------------------------------------
__global__ void k_rank(const float* __restrict__ scores,
                       int* __restrict__ idx_strong,
                       int* __restrict__ idx_weak) {
  const int b = blockIdx.x;
  const int n = threadIdx.x;
  __shared__ float s[NTOK];
  s[n] = scores[b * NTOK + n];
  __syncthreads();
  const float v = s[n];
  int rd = 0, ra = 0;
  for (int m = 0; m < NTOK; ++m) {
    const float w = s[m];
    rd += (w > v) || (w == v && m < n);   // position in descending stable order
    ra += (w < v) || (w == v && m < n);   // position in ascending  stable order
  }
  if (rd < KSTRONG) idx_strong[b * KSTRONG + rd] = n;
  if (ra < MWEAK)   idx_weak[b * MWEAK + ra] = n;
}

// ---------------------------------------------------------------------------
// Stage one raw (unscaled) strong 16 x 768 tile into an LDS buffer.
// Async path: 96 GLOBAL_LOAD_ASYNC_TO_LDS_B128 (512B per op, ASYNCcnt),
// lane -> half-row so each lane's strong index is loaded once.
// ---------------------------------------------------------------------------
__device__ __forceinline__ void stage_strong_tile(const float* __restrict__ imgb,
                                                  const int* __restrict__ idxs,
                                                  float* __restrict__ buf,
                                                  int lane, int kt) {
#if HAVE_ASYNC_LDS
  const int r = lane >> 1;                         // 0..15
  int kg = kt * 16 + r;
  if (kg > KSTRONG - 1) kg = KSTRONG - 1;          // clamp (masked at update)
  const float* src = imgb + (size_t)idxs[kg] * HDIM + (lane & 1) * (HDIM / 2);
  float*       dst = buf + r * LDSTRIDE + (lane & 1) * (HDIM / 2);
#pragma unroll 8
  for (int i = 0; i < HDIM / 2; i += 4) {
    __builtin_amdgcn_global_load_async_to_lds_b128(
        (GLOBAL_AS v4i*)(src + i), (LDS_AS v4i*)(dst + i), 0, 0);
  }
#else
  for (int i = lane; i < 16 * (HDIM / 4); i += 32) {
    const int r = i / (HDIM / 4);
    const int c = (i % (HDIM / 4)) * 4;
    int kg = kt * 16 + r;
    if (kg > KSTRONG - 1) kg = KSTRONG - 1;
    const float4 x = *(const float4*)(imgb + (size_t)idxs[kg] * HDIM + c);
    *(float4*)&buf[r * LDSTRIDE + c] = x;
  }
#endif
}

// ---------------------------------------------------------------------------
// K6: fused cosine-sim GEMM + row argmax via V_WMMA_F32_16X16X4_F32.
//     sim[m,k] = (rn_w[m]*v_weak[m]) . v_strong[k] * rn_s[k]; never stored.
//     Strong tiles double-buffered with async global->LDS prefetch.
// grid (11, 64), block 32 (one wave; EXEC all-1s at every WMMA).
// dynamic LDS: 3 tiles of 16 x LDSTRIDE floats (~145 KB, within 320KB/WGP)
// ---------------------------------------------------------------------------
__global__ void __launch_bounds__(32)
k_sim_argmax(const float* __restrict__ image,
             const float* __restrict__ rnorm,
             const int* __restrict__ idx_strong,
             const int* __restrict__ idx_weak,
             int* __restrict__ assign,
             float* __restrict__ theta_weak) {
  extern __shared__ float lds[];
  float* ldsA  = lds;                      // weak tile (pre-scaled) 16 x LDSTRIDE
  float* ldsB0 = lds + 16 * LDSTRIDE;      // strong tile ping (raw)
  float* ldsB1 = lds + 32 * LDSTRIDE;      // strong tile pong (raw)
  __shared__ int   rowIdx[16];
  __shared__ float rowScl[16];

  const int mt   = blockIdx.x;             // weak 16-row tile
  const int b    = blockIdx.y;
  const int lane = threadIdx.x;
  const int half = lane >> 4;              // 0: K{0,1} | 1: K{2,3}
  const int qq   = lane & 15;

  const float* imgb   = image + (size_t)b * NTOK * HDIM;
  const float* rnormb = rnorm + b * NTOK;
  const int*   idxsb  = idx_strong + b * KSTRONG;

  // stage weak row indices + normalization scales (zero padded rows)
  if (lane < 16) {
    const int m  = mt * 16 + lane;
    const int mi = (m < MWEAK) ? m : (MWEAK - 1);
    const int iw = idx_weak[b * MWEAK + mi];
    rowIdx[lane] = iw;
    rowScl[lane] = (m < MWEAK) ? rnormb[iw] : 0.0f;
  }
  __syncthreads();

  // load + normalize A (weak) tile into LDS (once per block)
  for (int i = lane; i < 16 * (HDIM / 4); i += 32) {
    const int r = i / (HDIM / 4);
    const int c = (i % (HDIM / 4)) * 4;
    const float  scl = rowScl[r];
    const float4 x   = *(const float4*)(imgb + (size_t)rowIdx[r] * HDIM + c);
    float* dst = &ldsA[r * LDSTRIDE + c];
    dst[0] = x.x * scl; dst[1] = x.y * scl; dst[2] = x.z * scl; dst[3] = x.w * scl;
  }

  float mx[8]; int mi8[8];
#pragma unroll
  for (int v = 0; v < 8; ++v) { mx[v] = -INFINITY; mi8[v] = 0; }

  // prefetch first strong tile
  stage_strong_tile(imgb, idxsb, ldsB0, lane, 0);

  for (int kt = 0; kt < KTILES; ++kt) {
    float* cur = (kt & 1) ? ldsB1 : ldsB0;
    float* nxt = (kt & 1) ? ldsB0 : ldsB1;

    // per-lane strong column scale for this tile (issued early to hide latency)
    const int  kg  = kt * 16 + qq;
    const int  kgc = (kg < KSTRONG) ? kg : (KSTRONG - 1);
    const float rnS = rnormb[idxsb[kgc]];

    if (kt + 1 < KTILES) {
      stage_strong_tile(imgb, idxsb, nxt, lane, kt + 1);  // overlap with compute
      wait_async_le96();   // in-order completion => tile kt resident in LDS
    } else {
      wait_async_0();
    }
    __syncthreads();

    // 16x16 sim tile: 192 rank-4 WMMA updates split over 4 accumulators
    v8f c0 = {}, c1 = {}, c2 = {}, c3 = {};
    const float* pa = &ldsA[qq * LDSTRIDE + 2 * half];
    const float* pb = &cur[qq * LDSTRIDE + 2 * half];
#pragma unroll 4
    for (int h0 = 0; h0 < HDIM; h0 += 16) {
      c0 = __builtin_amdgcn_wmma_f32_16x16x4_f32(
          false, *(const v2f*)(pa + h0 +  0), false, *(const v2f*)(pb + h0 +  0),
          (short)0, c0, false, false);
      c1 = __builtin_amdgcn_wmma_f32_16x16x4_f32(
          false, *(const v2f*)(pa + h0 +  4), false, *(const v2f*)(pb + h0 +  4),
          (short)0, c1, false, false);
      c2 = __builtin_amdgcn_wmma_f32_16x16x4_f32(
          false, *(const v2f*)(pa + h0 +  8), false, *(const v2f*)(pb + h0 +  8),
          (short)0, c2, false, false);
      c3 = __builtin_amdgcn_wmma_f32_16x16x4_f32(
          false, *(const v2f*)(pa + h0 + 12), false, *(const v2f*)(pb + h0 + 12),
          (short)0, c3, false, false);
    }
    const v8f cc = (c0 + c1) + (c2 + c3);

    // running argmax; C layout: VGPR v, lanes 0-15 -> row v, lanes 16-31 -> row 8+v
    if (kg < KSTRONG) {
#pragma unroll
      for (int v = 0; v < 8; ++v) {
        const float cv = cc[v] * rnS;                  // rank-1 column post-scale
        if (cv > mx[v]) { mx[v] = cv; mi8[v] = kg; }   // strict > keeps first idx
      }
    }
    __syncthreads();
  }

  // butterfly reduce across the 16 lanes holding each row (tie -> lower idx)
#pragma unroll
  for (int v = 0; v < 8; ++v) {
    float val = mx[v];
    int   idx = mi8[v];
#pragma unroll
    for (int d = 8; d >= 1; d >>= 1) {
      const float ov = __shfl_xor(val, d, 32);
      const int   oi = __shfl_xor(idx, d, 32);
      if (ov > val || (ov == val && oi < idx)) { val = ov; idx = oi; }
    }
    if (qq == 0) {
      const int m = mt * 16 + v + 8 * half;
      if (m < MWEAK) {
        assign[b * MWEAK + m] = idx;
        const float e = expf(val);
        theta_weak[b * MWEAK + m] = e / (e + 2.71828182845904523536f);
      }
    }
  }
}

// ---------------------------------------------------------------------------
// K7: segment-max (attended, theta_strong) + final blend, in-place in d_out.
//     Thread h owns column h of every row -> no inter-thread races on out.
// grid 64, block 768
// ---------------------------------------------------------------------------
__global__ void k_finalize(const float* __restrict__ image,
                           const int* __restrict__ idx_strong,
                           const int* __restrict__ idx_weak,
                           const int* __restrict__ assign,
                           const float* __restrict__ theta_weak,
                           float* __restrict__ out) {
  const int b = blockIdx.x;
  const int h = threadIdx.x;
  __shared__ float ts[KSTRONG];
  __shared__ int   valid[KSTRONG];
  __shared__ int   isb[KSTRONG];
  __shared__ int   asg[MWEAK];
  __shared__ int   iwb[MWEAK];
  __shared__ float tw[MWEAK];

  if (h < KSTRONG) {
    ts[h] = -INFINITY; valid[h] = 0;
    isb[h] = idx_strong[b * KSTRONG + h];
  }
  if (h < MWEAK) {
    asg[h] = assign[b * MWEAK + h];
    iwb[h] = idx_weak[b * MWEAK + h];
    tw[h]  = theta_weak[b * MWEAK + h];
  }
  __syncthreads();
  if (h == 0) {
    for (int m = 0; m < MWEAK; ++m) {
      const int k = asg[m];
      valid[k] = 1;
      ts[k] = fmaxf(ts[k], tw[m]);      // theta_strong = segment_max(theta_weak)
    }
  }
  __syncthreads();

  const float* imgb = image + (size_t)b * NTOK * HDIM;
  float* outb = out + (size_t)b * KSTRONG * HDIM;

  // attended = segment_max(v_weak) accumulated in out (column-owned, race-free)
  for (int r = 0; r < KSTRONG; ++r)
    outb[(size_t)r * HDIM + h] = -INFINITY;
  for (int m = 0; m < MWEAK; ++m) {
    const float xv = imgb[(size_t)iwb[m] * HDIM + h];
    float* p = &outb[(size_t)asg[m] * HDIM + h];
    *p = fmaxf(*p, xv);
  }
  // out = (1-theta)*v_strong + theta*attended  (invalid segments -> theta=0)
  for (int r = 0; r < KSTRONG; ++r) {
    const float vs  = imgb[(size_t)isb[r] * HDIM + h];
    const float th  = valid[r] ? ts[r] : 0.f;
    const float att = valid[r] ? outb[(size_t)r * HDIM + h] : 0.f;
    outb[(size_t)r * HDIM + h] = (1.f - th) * vs + th * att;
  }
}

// ---------------------------------------------------------------------------
extern "C" void kernel_launch(void* const* d_in, const int* in_sizes, int n_in,
                              void* d_out, int out_size, void* d_ws, size_t ws_size,
                              hipStream_t stream) {
  const float* text  = (const float*)d_in[0];   // (64,128,768)
  const float* image = (const float*)d_in[1];   // (64,576,768)
  const float* wqx   = (const float*)d_in[2];   // (12,768,64)
  const float* wkx   = (const float*)d_in[3];   // (12,768,64)
  float* out = (float*)d_out;                   // (64,403,768)

  // workspace layout (floats) — ~4.7 MB total
  float* ws         = (float*)d_ws;
  float* u          = ws;                                   // 12*64*768
  float* P          = u + NHEAD * BDIM * HDIM;              // 12*64*576
  float* rnorm      = P + NHEAD * BDIM * NTOK;              // 64*576
  float* scores     = rnorm + BDIM * NTOK;                  // 64*576
  int*   idx_strong = (int*)(scores + BDIM * NTOK);         // 64*403
  int*   idx_weak   = idx_strong + BDIM * KSTRONG;          // 64*173
  int*   assign     = idx_weak + BDIM * MWEAK;              // 64*173
  float* theta_weak = (float*)(assign + BDIM * MWEAK);      // 64*173

  k_compute_u  <<<dim3(NHEAD, BDIM), 64,  0, stream>>>(text, wqx, wkx, u);
  k_score_rnorm<<<BDIM, 256, 0, stream>>>(image, u, P, rnorm);
  k_softmax    <<<NHEAD * BDIM, 256, 0, stream>>>(P);
  k_mean       <<<BDIM, NTOK, 0, stream>>>(P, scores);
  k_rank       <<<BDIM, NTOK, 0, stream>>>(scores, idx_strong, idx_weak);

  const size_t shmem = (size_t)3 * 16 * LDSTRIDE * sizeof(float);  // ~145 KB
  k_sim_argmax <<<dim3(MTILES, BDIM), 32, shmem, stream>>>(
      image, rnorm, idx_strong, idx_weak, assign, theta_weak);

  k_finalize   <<<BDIM, HDIM, 0, stream>>>(image, idx_strong, idx_weak,
                                           assign, theta_weak, out);
}